// TrainablePCEN_42107859370067
// MI455X (gfx1250) — compile-verified
//
#include <hip/hip_runtime.h>
#include <stdint.h>

// ---------------- problem constants ----------------
#define TPB        256        // 8 waves (wave32)
#define T_LEN      4096       // time dimension / row length
#define CHUNK      16         // elements per thread
#define PAD_STRIDE 17         // 16 data floats + 1 TDM pad float per chunk
#define S_COEF     0.025f
#define A_COEF     0.975f     // 1 - S
#define EPSV       1e-6f

typedef unsigned int u32x4 __attribute__((ext_vector_type(4)));
typedef int          i32x4 __attribute__((ext_vector_type(4)));
typedef int          i32x8 __attribute__((ext_vector_type(8)));
typedef float        f32x4 __attribute__((ext_vector_type(4)));

// raw hardware base-2 transcendentals (inputs are guaranteed normal-range here)
__device__ __forceinline__ float hw_log2(float v) { return __builtin_amdgcn_logf(v); }
__device__ __forceinline__ float hw_exp2(float v) { return __builtin_amdgcn_exp2f(v); }

// b^e for e <= 31, fixed trip count (no lane divergence)
__device__ __forceinline__ float powu5(float b, unsigned e) {
    float r = 1.0f, p = b;
#pragma unroll
    for (int i = 0; i < 5; ++i) {
        r = (e & 1u) ? r * p : r;
        p *= p;
        e >>= 1u;
    }
    return r;
}

__global__ __launch_bounds__(TPB) void pcen_tdm_kernel(
        const float* __restrict__ x,
        const float* __restrict__ p_log_alpha,
        const float* __restrict__ p_log_delta,
        const float* __restrict__ p_log_r,
        float* __restrict__ out) {
    // TDM-padded row staging: 4096 data DWORDs + 1 pad DWORD per 16 -> 4352 floats
    __shared__ float smem[T_LEN + T_LEN / CHUNK];
    __shared__ float waveTot[TPB / 32];

    const int tid  = threadIdx.x;
    const int lane = tid & 31;
    const int wav  = tid >> 5;
    const long long base = (long long)blockIdx.x * T_LEN;

    // -------- 1) async DMA: global row -> LDS via Tensor Data Mover --------
#if __has_builtin(__builtin_amdgcn_tensor_load_to_lds)
    {
        const uint64_t ga      = (uint64_t)(uintptr_t)(const void*)(x + base);
        const uint32_t lds_off = (uint32_t)(uintptr_t)(&smem[0]); // low 32 bits of flat LDS ptr = group-relative byte offset

        // D# group 0: count=1 | lds_addr | global_addr[56:0] | type=2 ("image")
        u32x4 g0;
        g0.x = 1u;
        g0.y = lds_off;
        g0.z = (uint32_t)ga;
        g0.w = ((uint32_t)(ga >> 32) & 0x01FFFFFFu) | 0x80000000u;

        // D# group 1:
        //  word0: data_size=2 (4B)<<16 | pad_enable<<20 | pad_interval=3 (16 DW)<<22 | pad_amount=0 (1 DW)<<25
        //  tensor_dim0 = 4096 (bits 79:48), tensor_dim1 = 1 (bits 111:80)
        //  tile_dim0 = 4096 (bits 127:112), tile_dim1 = 1 (bits 143:128), tile_dim2 = 0
        //  tensor_dim0_stride = 4096 (bits 207:160)
        i32x8 g1;
        g1[0] = 0x00D20000;            // data_size | pad_enable | pad_interval=16DW | pad_amount=1DW
        g1[1] = (int)(T_LEN << 16);    // tensor_dim0[15:0] in bits 63:48
        g1[2] = 0x00010000;            // tensor_dim0 hi = 0, tensor_dim1 = 1
        g1[3] = (int)(T_LEN << 16);    // tile_dim0 = 4096 in bits 127:112
        g1[4] = 0x00000001;            // tile_dim1 = 1, tile_dim2 = 0
        g1[5] = T_LEN;                 // tensor_dim0_stride (low 32)
        g1[6] = 0;
        g1[7] = 0;

        i32x4 gz = {0, 0, 0, 0};

        if (wav == 0) {                // one DMA per workgroup (TDM ignores EXEC)
#if __clang_major__ >= 23
            i32x8 gz8 = {0, 0, 0, 0, 0, 0, 0, 0};
            __builtin_amdgcn_tensor_load_to_lds(g0, g1, gz, gz, gz8, 0);
#else
            __builtin_amdgcn_tensor_load_to_lds(g0, g1, gz, gz, 0);
#endif
            __builtin_amdgcn_s_wait_tensorcnt(0);
        }
    }
#else
    // fallback: coalesced manual load with the same pad layout
    for (int i = tid; i < T_LEN; i += TPB) smem[i + (i >> 4)] = x[base + i];
#endif
    __syncthreads();

    // -------- 2) load chunk from LDS (stride-17 => bank-conflict-free) --------
    float xv[CHUNK];
    const int sbase = tid * PAD_STRIDE;
#pragma unroll
    for (int k = 0; k < CHUNK; ++k) xv[k] = smem[sbase + k];

    // local EMA with zero carry-in -> chunk total
    float m = 0.0f;
#pragma unroll
    for (int k = 0; k < CHUNK; ++k) m = A_COEF * m + S_COEF * xv[k];
    float c = m;

    // decay constants: q = a^16 (per chunk), Q = q^32 (per wave)
    const float a2 = A_COEF * A_COEF, a4 = a2 * a2, a8 = a4 * a4;
    const float q  = a8 * a8;                       // a^16
    const float q2 = q * q, q4 = q2 * q2, q8 = q4 * q4, q16 = q8 * q8;
    const float Q  = q16 * q16;                     // a^512

    // -------- 3) wave32 Kogge-Stone scan of the linear recurrence c_i = t_i + q*c_{i-1}
    float f = q;
#pragma unroll
    for (int d = 1; d < 32; d <<= 1) {
        float up = __shfl_up(c, (unsigned)d, 32);
        if (lane >= d) c += f * up;
        f *= f;
    }
    if (lane == 31) waveTot[wav] = c;               // this wave's total (zero carry-in)
    __syncthreads();

    // combine across the 8 waves: waveIn_w = t_{w-1} + Q * waveIn_{w-1}
    float waveIn = 0.0f;
    for (int v = 0; v < wav; ++v) waveIn = waveTot[v] + Q * waveIn;

    float excl = __shfl_up(c, 1u, 32);              // exclusive within-wave prefix
    if (lane == 0) excl = 0.0f;
    const float m_in = excl + powu5(q, (unsigned)lane) * waveIn; // true carry into this chunk

    // -------- 4) PCEN pointwise math + second EMA pass with true carry --------
    // All pow() in native base-2: v_log_f32 is log2, v_exp_f32 is exp2.
    //   x/(EPS+m)^a  == x * exp2(-a * log2(EPS+m))
    //   v^r          == exp2(r * log2(v))
    // Inputs are normal-range (EPS+m >= 1e-6, v >= delta), so no denorm fixup needed.
    const float alpha  = __expf(p_log_alpha[0]) * 0.98f;
    const float delta  = __expf(p_log_delta[0]) * 2.0f;
    const float r      = __expf(p_log_r[0]) * 0.5f;
    const float dr     = hw_exp2(r * hw_log2(delta));  // delta^r
    const float nalpha = -alpha;

    float res[CHUNK];
    float mm = m_in;
#pragma unroll
    for (int k = 0; k < CHUNK; ++k) {
        mm = A_COEF * mm + S_COEF * xv[k];
        const float inv_denom = hw_exp2(nalpha * hw_log2(EPSV + mm)); // 1/(EPS+m)^alpha
        const float v         = __fmaf_rn(xv[k], inv_denom, delta);
        res[k] = hw_exp2(r * hw_log2(v)) - dr;
    }

    // -------- 5) streaming non-temporal 128-bit stores --------
    float* orow = out + base + (long long)tid * CHUNK;
#pragma unroll
    for (int j = 0; j < CHUNK / 4; ++j) {
        f32x4 o = { res[4 * j], res[4 * j + 1], res[4 * j + 2], res[4 * j + 3] };
        __builtin_nontemporal_store(o, (f32x4*)(orow + 4 * j));
    }
}

extern "C" void kernel_launch(void* const* d_in, const int* in_sizes, int n_in,
                              void* d_out, int out_size, void* d_ws, size_t ws_size,
                              hipStream_t stream) {
    const float* x  = (const float*)d_in[0];
    const float* la = (const float*)d_in[1];
    const float* ld = (const float*)d_in[2];
    const float* lr = (const float*)d_in[3];
    float* out = (float*)d_out;

    const int rows = in_sizes[0] / T_LEN;   // 64 * 128 = 8192 independent scan rows
    pcen_tdm_kernel<<<rows, TPB, 0, stream>>>(x, la, ld, lr, out);
}